// I5Pool_326417514937
// MI455X (gfx1250) — compile-verified
//
#include <hip/hip_runtime.h>

// Problem constants from the reference: x[8,256,128,128] f32, guide[8,1,128,128] f32
#define B_ 8
#define C_ 256
#define H_ 128
#define W_ 128

// gfx1250 async global->LDS path (ASYNCcnt), guarded so a signature/name
// mismatch degrades to a plain cooperative LDS copy instead of failing.
#if defined(__has_builtin)
#if __has_builtin(__builtin_amdgcn_global_load_async_to_lds_b128) && \
    __has_builtin(__builtin_amdgcn_s_wait_asynccnt)
#define USE_ASYNC_LDS 1
#endif
#endif
#ifndef USE_ASYNC_LDS
#define USE_ASYNC_LDS 0
#endif

// Builtin signature (from hipcc diagnostic): param 1 is
//   '__attribute__((__vector_size__(4*sizeof(int)))) int __device__ *'
// i.e. int4 in address_space(1) (HIP prints AS1 as __device__), and by
// symmetry the LDS destination is int4 in address_space(3).
typedef int v4i __attribute__((ext_vector_type(4)));
typedef __attribute__((address_space(1))) v4i gv4i;   // global
typedef __attribute__((address_space(3))) v4i lv4i;   // LDS

// wave32 shuffle-up via ds_bpermute_b32 (index is byte offset = lane*4)
__device__ __forceinline__ float shfl_up_f32(float v, int lane, int delta) {
  int src = lane - delta;
  if (src < 0) src = lane;              // value unused when lane < delta
  union { float f; int i; } u; u.f = v;
  u.i = __builtin_amdgcn_ds_bpermute(src << 2, u.i);
  return u.f;
}

__global__ __launch_bounds__(256)
void I5Pool_gated_cummax_kernel(const float* __restrict__ x,
                                const float* __restrict__ guide,
                                float* __restrict__ out) {
  __shared__ float sg[W_];              // one guide row, shared by 8 waves

  // blockIdx.x = (b*H + h) * 8 + channel_group
  const int cg = blockIdx.x & 7;        // 8 groups of 32 channels
  const int bh = blockIdx.x >> 3;       // b*H + h
  const int h  = bh & (H_ - 1);
  const int b  = bh >> 7;               // / 128

  const int tid  = threadIdx.x;
  const int lane = tid & 31;
  const int wave = tid >> 5;

  const long long goff = (long long)bh * W_;

#if USE_ASYNC_LDS
  if (wave == 0) {
    // 32 lanes x 16B = whole 512B guide row, DMA'd to LDS (ASYNCcnt path)
    const float* gsrc = guide + goff + (lane << 2);
    float*       ldst = &sg[lane << 2];
    __builtin_amdgcn_global_load_async_to_lds_b128((gv4i*)gsrc,
                                                   (lv4i*)ldst,
                                                   /*offset=*/0, /*cpol=*/0);
    __builtin_amdgcn_s_wait_asynccnt(0);
  }
#else
  if (tid < W_) sg[tid] = guide[goff + tid];
#endif
  __syncthreads();

  const float4 g4 = *(const float4*)&sg[lane << 2];
  const float NEG_INF = -__builtin_inff();

  const int c0 = (cg << 5) + (wave << 2);   // 4 channels per wave

#pragma unroll
  for (int i = 0; i < 4; ++i) {
    const int c = c0 + i;
    const long long roff =
        ((((long long)b * C_ + c) * H_ + h) * (long long)W_) + (lane << 2);

    // coalesced 128-bit load of 4 gated elements per lane
    float4 v = *(const float4*)(x + roff);
    const float e0 = v.x * g4.x;
    const float e1 = v.y * g4.y;
    const float e2 = v.z * g4.z;
    const float e3 = v.w * g4.w;

    // in-lane inclusive cummax of the 4 elements
    const float m0 = e0;
    const float m1 = fmaxf(m0, e1);
    const float m2 = fmaxf(m1, e2);
    const float m3 = fmaxf(m2, e3);

    // wave32 inclusive scan (max) of per-lane totals: 5 bpermute steps
    float s = m3;
    #pragma unroll
    for (int off = 1; off < 32; off <<= 1) {
      const float t = shfl_up_f32(s, lane, off);
      if (lane >= off) s = fmaxf(s, t);
    }
    // exclusive prefix for this lane
    float p = shfl_up_f32(s, lane, 1);
    if (lane == 0) p = NEG_INF;

    float4 o;
    o.x = fmaxf(p, m0);
    o.y = fmaxf(p, m1);
    o.z = fmaxf(p, m2);
    o.w = fmaxf(p, m3);

    *(float4*)(out + roff) = o;   // coalesced 128-bit store
  }
}

extern "C" void kernel_launch(void* const* d_in, const int* in_sizes, int n_in,
                              void* d_out, int out_size, void* d_ws, size_t ws_size,
                              hipStream_t stream) {
  const float* x     = (const float*)d_in[0];
  const float* guide = (const float*)d_in[1];
  float*       out   = (float*)d_out;

  (void)in_sizes; (void)n_in; (void)out_size; (void)d_ws; (void)ws_size;

  dim3 grid(B_ * H_ * (C_ / 32));   // 8*128*8 = 8192 blocks
  dim3 block(256);                  // 8 wave32 waves per block
  hipLaunchKernelGGL(I5Pool_gated_cummax_kernel, grid, block, 0, stream,
                     x, guide, out);
}